// SwitchGNN_41807211660046
// MI455X (gfx1250) — compile-verified
//
#include <hip/hip_runtime.h>
#include <hip/hip_bf16.h>

typedef __attribute__((ext_vector_type(2))) float v2f;
typedef __attribute__((ext_vector_type(8))) float v8f;

#define GNN_N 100000
#define GNN_D 128
#define GNN_E 400000
#define GNN_T 7

// One wave per edge: 32 lanes x float4 = 128 features, coalesced gather from
// L2-resident x, hardware f32 atomics into the per-type accumulator.
__global__ void gnn_scatter_kernel(const float* __restrict__ x,
                                   const int* __restrict__ src,
                                   const int* __restrict__ dst,
                                   float* __restrict__ summed,
                                   float* __restrict__ cnt,
                                   int n_edges) {
    int tid  = blockIdx.x * blockDim.x + threadIdx.x;
    int e    = tid >> 5;
    int lane = tid & 31;
    if (e >= n_edges) return;
    int s = src[e];
    int d = dst[e];
    const float4 val = *(const float4*)(x + (size_t)s * GNN_D + lane * 4);
    float* p = summed + (size_t)d * GNN_D + lane * 4;
    unsafeAtomicAdd(p + 0, val.x);
    unsafeAtomicAdd(p + 1, val.y);
    unsafeAtomicAdd(p + 2, val.z);
    unsafeAtomicAdd(p + 3, val.w);
    if (lane == 0) unsafeAtomicAdd(cnt + d, 1.0f);
}

// One wave computes a 16-row x 128-col strip of (mean @ W_t) via
// V_WMMA_F32_16X16X4_F32 (full fp32 matrix path), K-loop of 32 steps,
// 8 N-tiles -> 256 wmma per wave. Accumulates scaled result into out.
__global__ void gnn_gemm_acc_kernel(const float* __restrict__ summed, // [N,128]
                                    const float* __restrict__ cnt,    // [N]
                                    const float* __restrict__ W,      // [128,128] row-major (k, n)
                                    const float* __restrict__ ball,   // [7,128] all biases
                                    float* __restrict__ out,          // [N,128]
                                    int n_tiles, int is_first) {
    int wave = (blockIdx.x * blockDim.x + threadIdx.x) >> 5;
    int lane = threadIdx.x & 31;
    if (wave >= n_tiles) return;   // wave-uniform: EXEC stays all-ones for WMMA

    const int m0   = wave * 16;
    const int half = lane >> 4;    // 0: K=2h+v low pair, 1: high pair
    const int l16  = lane & 15;

    // A-matrix 16x4 f32 layout: lane l16 holds row M=m0+l16, K = 2*half + {0,1}
    const int arow = m0 + l16;
    float ic = cnt[arow];
    ic = 1.0f / fmaxf(ic, 1.0f);
    const float* arowp = summed + (size_t)arow * GNN_D + half * 2;

    v8f c[8];
#pragma unroll
    for (int i = 0; i < 8; ++i) c[i] = (v8f)(0.0f);

    for (int kb = 0; kb < GNN_D; kb += 4) {
        v2f a = *(const v2f*)(arowp + kb);
        a.x *= ic;
        a.y *= ic;
        const int k0 = kb + half * 2;
#pragma unroll
        for (int nt = 0; nt < 8; ++nt) {
            // B-matrix 4x16 layout: VGPR v holds W[k0+v][n0+l16]
            const int n = nt * 16 + l16;
            v2f bfrag;
            bfrag.x = W[(size_t)(k0 + 0) * GNN_D + n];
            bfrag.y = W[(size_t)(k0 + 1) * GNN_D + n];
            c[nt] = __builtin_amdgcn_wmma_f32_16x16x4_f32(
                false, a, false, bfrag, (short)0, c[nt], false, false);
        }
    }

    const float scale = 1.0f / (float)GNN_T;
#pragma unroll
    for (int nt = 0; nt < 8; ++nt) {
        const int n = nt * 16 + l16;
        float binit = 0.0f;
        if (is_first) {
#pragma unroll
            for (int t = 0; t < GNN_T; ++t) binit += ball[t * GNN_D + n];
            binit *= scale;
        }
        // C/D layout: VGPR v -> row m0 + v + 8*half, col n
#pragma unroll
        for (int v = 0; v < 8; ++v) {
            const size_t o = (size_t)(m0 + v + half * 8) * GNN_D + n;
            const float add = c[nt][v] * scale;
            if (is_first) out[o] = add + binit;
            else          out[o] += add;
        }
    }
}

extern "C" void kernel_launch(void* const* d_in, const int* in_sizes, int n_in,
                              void* d_out, int out_size, void* d_ws, size_t ws_size,
                              hipStream_t stream) {
    const float* x     = (const float*)d_in[0];
    const int*   edges = (const int*)d_in[1];   // [7, 2, E]
    const float* W     = (const float*)d_in[2]; // [7, 128, 128]
    const float* b     = (const float*)d_in[3]; // [7, 128]
    float* out = (float*)d_out;

    float* summed = (float*)d_ws;                       // [N, 128]
    float* cnt    = summed + (size_t)GNN_N * GNN_D;     // [N]
    const size_t scratch_bytes = ((size_t)GNN_N * GNN_D + GNN_N) * sizeof(float);

    const int scatter_threads = GNN_E * 32;
    const int scatter_blocks  = (scatter_threads + 255) / 256;
    const int n_tiles     = GNN_N / 16;                  // 6250
    const int gemm_blocks = (n_tiles + 7) / 8;           // 8 waves per 256-thread block

    for (int t = 0; t < GNN_T; ++t) {
        hipMemsetAsync(summed, 0, scratch_bytes, stream);
        const int* src = edges + (size_t)(t * 2 + 0) * GNN_E;
        const int* dst = edges + (size_t)(t * 2 + 1) * GNN_E;
        gnn_scatter_kernel<<<scatter_blocks, 256, 0, stream>>>(x, src, dst, summed, cnt, GNN_E);
        gnn_gemm_acc_kernel<<<gemm_blocks, 256, 0, stream>>>(
            summed, cnt, W + (size_t)t * GNN_D * GNN_D, b, out, n_tiles, t == 0);
    }
}